// GCN_64785286693080
// MI455X (gfx1250) — compile-verified
//
#include <hip/hip_runtime.h>
#include <math.h>

// ---------------------------------------------------------------------------
// GCN 2-layer forward for MI455X (gfx1250, wave32).
//
// Bandwidth roofline (~3 GB @ 23.3 TB/s => ~130us floor):
//   1. deg/dinv:      deg = 1 (self loop) + f32 atomic counts over dst; rsqrt
//   2. xw = x @ W1:   f32 WMMA 16x16x4, one wave per 16-node tile (K=100=25*4)
//   3. h   = scatter-add( dinv[s]*dinv[d] * xw[s] ) over E+N edges (16 floats)
//   4. h   = relu(h + b1)
//   5. a2  = scatter-add( dinv[s]*dinv[d] * h[s] )  (16 floats; reuses xw buf)
//   6. out = log_softmax(a2 @ W2 + b2)              (per-node, 16->20)
//
// L2 policy: edge_index (128MB, streamed 3x) is loaded non-temporal so the
// 192MB L2 stays resident with the gather/scatter feature tables (xw/h, 32MB
// each, ~17x reuse) and dinv (2MB). Final output is stored non-temporal.
// ---------------------------------------------------------------------------

typedef __attribute__((ext_vector_type(2))) float v2f;
typedef __attribute__((ext_vector_type(8))) float v8f;

#define DIN  100
#define HID  16
#define NCLS 20

__global__ void fill_f32(float* __restrict__ p, float v, int n) {
    int i = blockIdx.x * blockDim.x + threadIdx.x;
    if (i < n) p[i] = v;
}

__global__ void deg_accum(const long long* __restrict__ dst, float* __restrict__ deg, int E) {
    int i = blockIdx.x * blockDim.x + threadIdx.x;
    if (i < E) {
        int d = (int)__builtin_nontemporal_load(&dst[i]);   // streaming read
        atomicAdd(&deg[d], 1.0f);
    }
}

__global__ void rsqrt_inplace(float* __restrict__ deg, int n) {
    int i = blockIdx.x * blockDim.x + threadIdx.x;
    if (i < n) {
        float d = deg[i];
        deg[i] = (d > 0.0f) ? rsqrtf(d) : 0.0f;
    }
}

// xw[N,16] = x[N,100] @ W1[100,16] using V_WMMA_F32_16X16X4_F32.
// One wave computes one 16x16 output tile (16 nodes x 16 hidden).
// A 16x4 f32 fragment: lane m=lane&15 is row M; VGPR g holds K = g + 2*(lane>>4).
// B 4x16 fragment is symmetric (lane&15 = column N).
// C/D 16x16 f32: VGPR v holds row M = v + 8*(lane>>4), col N = lane&15.
__global__ void gemm_x_w1(const float* __restrict__ x, const float* __restrict__ W1,
                          float* __restrict__ xw, int N) {
    int wave = threadIdx.x >> 5;
    int lane = threadIdx.x & 31;
    int tile = blockIdx.x * (blockDim.x >> 5) + wave;
    int row0 = tile * 16;
    if (row0 >= N) return;              // wave-uniform exit: EXEC all-ones for WMMA

    int hi = lane >> 4;                 // half-wave select
    int m  = lane & 15;                 // A row / B,C,D column

    int arow = row0 + m;
    if (arow >= N) arow = N - 1;        // clamp reads for a partial tail tile
    const float* rowp = x + (long)arow * DIN + 2 * hi;

    v8f acc = {0.f, 0.f, 0.f, 0.f, 0.f, 0.f, 0.f, 0.f};

#pragma unroll 5
    for (int k0 = 0; k0 < DIN; k0 += 4) {
        v2f a, b;
        a.x = rowp[k0];                         // x[arow, k0 + 2*hi]
        a.y = rowp[k0 + 1];                     // x[arow, k0 + 2*hi + 1]
        b.x = W1[(k0 + 2 * hi) * HID + m];      // W1[k0+2*hi, n]
        b.y = W1[(k0 + 2 * hi + 1) * HID + m];  // W1[k0+2*hi+1, n]
        acc = __builtin_amdgcn_wmma_f32_16x16x4_f32(
            /*neg_a=*/false, a, /*neg_b=*/false, b,
            /*c_mod=*/(short)0, acc, /*reuse_a=*/false, /*reuse_b=*/false);
    }

    if (row0 + 16 <= N) {
        // full tile (the only case for N % 16 == 0): branch-free, one base
        // address + 8 immediate-offset coalesced stores
        float* outp = xw + (long)row0 * HID + (long)hi * 8 * HID + m;
#pragma unroll
        for (int v = 0; v < 8; ++v) outp[(long)v * HID] = acc[v];
    } else {
        for (int v = 0; v < 8; ++v) {
            int orow = row0 + v + 8 * hi;
            if (orow < N) xw[(long)orow * HID + m] = acc[v];
        }
    }
}

// out[d] += dinv[s]*dinv[d] * feat[s]  over E edges + N self-loops, 16 feats/row.
__global__ void aggregate16(const float* __restrict__ feat,
                            const long long* __restrict__ src,
                            const long long* __restrict__ dst,
                            const float* __restrict__ dinv,
                            float* __restrict__ out, int E, int N) {
    int i = blockIdx.x * blockDim.x + threadIdx.x;
    if (i >= E + N) return;
    int s, d;
    if (i < E) {
        s = (int)__builtin_nontemporal_load(&src[i]);   // streaming reads
        d = (int)__builtin_nontemporal_load(&dst[i]);
    } else {
        s = d = i - E;                                  // self loop
    }
    float nrm = dinv[s] * dinv[d];
    const float4* fp = (const float4*)(feat + (long)s * HID);
    float* op = out + (long)d * HID;
#pragma unroll
    for (int j = 0; j < 4; ++j) {
        float4 v = fp[j];
        atomicAdd(op + 4 * j + 0, nrm * v.x);
        atomicAdd(op + 4 * j + 1, nrm * v.y);
        atomicAdd(op + 4 * j + 2, nrm * v.z);
        atomicAdd(op + 4 * j + 3, nrm * v.w);
    }
}

__global__ void relu_bias16(float* __restrict__ h, const float* __restrict__ b1, int total) {
    int i = blockIdx.x * blockDim.x + threadIdx.x;
    if (i < total) {
        float v = h[i] + b1[i & (HID - 1)];
        h[i] = v > 0.0f ? v : 0.0f;
    }
}

// out[N,20] = log_softmax(agg[N,16] @ W2[16,20] + b2)
__global__ void head_logsoftmax(const float* __restrict__ agg, const float* __restrict__ W2,
                                const float* __restrict__ b2, float* __restrict__ out, int N) {
    int i = blockIdx.x * blockDim.x + threadIdx.x;
    if (i >= N) return;
    float hv[HID];
    const float4* hp = (const float4*)(agg + (long)i * HID);
#pragma unroll
    for (int j = 0; j < 4; ++j) {
        float4 v = hp[j];
        hv[4 * j + 0] = v.x; hv[4 * j + 1] = v.y;
        hv[4 * j + 2] = v.z; hv[4 * j + 3] = v.w;
    }
    float logits[NCLS];
    float mx = -INFINITY;
#pragma unroll
    for (int c = 0; c < NCLS; ++c) {
        float a = b2[c];
#pragma unroll
        for (int k = 0; k < HID; ++k) a = fmaf(hv[k], W2[k * NCLS + c], a);
        logits[c] = a;
        mx = fmaxf(mx, a);
    }
    float sum = 0.0f;
#pragma unroll
    for (int c = 0; c < NCLS; ++c) sum += expf(logits[c] - mx);
    float lse = mx + logf(sum);
    float* op = out + (long)i * NCLS;
#pragma unroll
    for (int c = 0; c < NCLS; ++c)
        __builtin_nontemporal_store(logits[c] - lse, op + c);  // write-once output
}

extern "C" void kernel_launch(void* const* d_in, const int* in_sizes, int n_in,
                              void* d_out, int out_size, void* d_ws, size_t ws_size,
                              hipStream_t stream) {
    const float*     x  = (const float*)d_in[0];
    const long long* ei = (const long long*)d_in[1];
    const float*     W1 = (const float*)d_in[2];
    const float*     b1 = (const float*)d_in[3];
    const float*     W2 = (const float*)d_in[4];
    const float*     b2 = (const float*)d_in[5];

    int N = in_sizes[0] / DIN;    // 500000
    int E = in_sizes[1] / 2;      // 8000000
    const long long* srcv = ei;         // edge_index[0]
    const long long* dstv = ei + E;     // edge_index[1]

    // workspace: dinv[N] | xw[N*16] | h[N*16]   (~66 MB)
    float* dinv = (float*)d_ws;
    float* xw   = dinv + N;
    float* h    = xw + (size_t)N * HID;

    const int T = 256;
    int nf = N * HID;

    // 1) degree (seeded with self-loop) -> dinv
    fill_f32<<<(N + T - 1) / T, T, 0, stream>>>(dinv, 1.0f, N);
    deg_accum<<<(E + T - 1) / T, T, 0, stream>>>(dstv, dinv, E);
    rsqrt_inplace<<<(N + T - 1) / T, T, 0, stream>>>(dinv, N);

    // 2) xw = x @ W1 (WMMA f32 16x16x4)
    int tiles = (N + 15) / 16;
    gemm_x_w1<<<(tiles + 7) / 8, 256, 0, stream>>>(x, W1, xw, N);

    // 3) layer-1 aggregation
    fill_f32<<<(nf + T - 1) / T, T, 0, stream>>>(h, 0.0f, nf);
    aggregate16<<<((E + N) + T - 1) / T, T, 0, stream>>>(xw, srcv, dstv, dinv, h, E, N);

    // 4) bias + ReLU
    relu_bias16<<<(nf + T - 1) / T, T, 0, stream>>>(h, b1, nf);

    // 5) layer-2 aggregation (aggregate 16 feats first, transform after; reuse xw)
    fill_f32<<<(nf + T - 1) / T, T, 0, stream>>>(xw, 0.0f, nf);
    aggregate16<<<((E + N) + T - 1) / T, T, 0, stream>>>(h, srcv, dstv, dinv, xw, E, N);

    // 6) head: (agg2 @ W2 + b2) -> log_softmax
    head_logsoftmax<<<(N + T - 1) / T, T, 0, stream>>>(xw, W2, b2, (float*)d_out, N);
}